// High_enhancer_62801011802557
// MI455X (gfx1250) — compile-verified
//
#include <hip/hip_runtime.h>

typedef __attribute__((ext_vector_type(2))) float v2f;
typedef __attribute__((ext_vector_type(8))) float v8f;

#define C32 32

// ---------------- Phase 0: zero the pooled accumulator -------------------
__global__ void zero_f32_kernel(float* __restrict__ p, long n) {
    long i = (long)blockIdx.x * blockDim.x + threadIdx.x;
    long stride = (long)gridDim.x * blockDim.x;
    for (; i < n; i += stride) p[i] = 0.0f;
}

// ---------------- Phase 1: sparse avg-pool scatter-add -------------------
// thread t handles (pair = t/32, channel = t%32): coalesced 128B row gathers,
// atomic adds to consecutive addresses of the destination row.
__global__ void pool_scatter_kernel(const float* __restrict__ feat,
                                    const int* __restrict__ pin,
                                    const int* __restrict__ pout,
                                    float* __restrict__ pooled, long P32) {
    long t = (long)blockIdx.x * blockDim.x + threadIdx.x;
    if (t >= P32) return;
    long pair = t >> 5;
    int  c    = (int)(t & 31);
    float v = feat[(long)pin[pair] * C32 + c];
    atomicAdd(&pooled[(long)pout[pair] * C32 + c], v);
}

// ---------------- Phase 2: divide by true neighbor count -----------------
__global__ void pool_div_kernel(float* __restrict__ pooled,
                                const int* __restrict__ counts, long M32) {
    long t = (long)blockIdx.x * blockDim.x + threadIdx.x;
    if (t >= M32) return;
    int cnt = counts[t >> 5];
    pooled[t] = pooled[t] / (float)(cnt > 0 ? cnt : 1);
}

// ---------------- Phase 3: out = feat - bias -----------------------------
__global__ void init_out_kernel(const float* __restrict__ feat,
                                const float* __restrict__ bias,
                                float* __restrict__ out, long NC) {
    long t = (long)blockIdx.x * blockDim.x + threadIdx.x;
    if (t >= NC) return;
    out[t] = feat[t] - bias[t & 31];
}

// ---------------- Phase 4: WMMA transposed-conv scatter ------------------
// One wave32 per tile of 16 consecutive kernel-map pairs. up_k is
// non-decreasing, so a tile nearly always uses a single 32x32 weight:
//   OUT[16x32] = IN[16x32] x W[32x32]  via 16x v_wmma_f32_16x16x4_f32
// (8 K-steps of 4, 2 N-tiles of 16). Results scattered with -sign into out.
__global__ __launch_bounds__(256)
void upconv_wmma_kernel(const float* __restrict__ pooled,   // [M][32]
                        const float* __restrict__ weight,   // [27][32][32] (cin-major)
                        const int* __restrict__ up_in,
                        const int* __restrict__ up_out,
                        const int* __restrict__ up_k,
                        float* __restrict__ out,            // [N][32]
                        int Q) {
    const int lane = (int)(threadIdx.x & 31);
    const int wave = (int)(threadIdx.x >> 5);
    const int tile = (int)blockIdx.x * 8 + wave;
    const int q0   = tile * 16;
    if (q0 >= Q) return;
    const int rows = (Q - q0 < 16) ? (Q - q0) : 16;

    const int half = lane >> 4;   // 0: lanes 0-15, 1: lanes 16-31
    const int l16  = lane & 15;

    const int kFirst = up_k[q0];
    const int kLast  = up_k[q0 + rows - 1];

    if (rows == 16 && kFirst == kLast) {
        // ---- fast path: uniform weight across the tile ----
        const float* __restrict__ W = weight + (long)kFirst * C32 * C32;
        // A-matrix row for this lane (rows 0..15 duplicated across halves)
        const long arow = (long)up_in[q0 + l16];
        const float* __restrict__ inrow = pooled + arow * C32;

        #pragma unroll
        for (int ntile = 0; ntile < 2; ++ntile) {
            const int n0 = ntile * 16;
            v8f acc = {};
            #pragma unroll
            for (int kt = 0; kt < 8; ++kt) {
                const int kb = kt * 4 + half * 2;   // this half's K offsets
                v2f a;                               // A 16x4: lanes<16 K0,K1 ; lanes>=16 K2,K3
                a.x = inrow[kb + 0];
                a.y = inrow[kb + 1];
                v2f b;                               // B 4x16: VGPR0=K0/K2, VGPR1=K1/K3
                b.x = W[(long)(kb + 0) * C32 + n0 + l16];
                b.y = W[(long)(kb + 1) * C32 + n0 + l16];
                acc = __builtin_amdgcn_wmma_f32_16x16x4_f32(
                        /*neg_a=*/false, a, /*neg_b=*/false, b,
                        /*c_mod=*/(short)0, acc,
                        /*reuse_a=*/false, /*reuse_b=*/false);
            }
            // D 16x16: VGPR j holds row (j + 8*half), col (n0 + l16)
            #pragma unroll
            for (int j = 0; j < 8; ++j) {
                const int m    = j + 8 * half;
                const long orow = (long)up_out[q0 + m];
                atomicAdd(&out[orow * C32 + n0 + l16], -acc[j]);
            }
        }
    } else {
        // ---- rare slow path: ragged tail tile or weight-boundary tile ----
        for (int r = 0; r < rows; ++r) {
            const int q = q0 + r;
            const float* __restrict__ inrow = pooled + (long)up_in[q] * C32;
            const float* __restrict__ W     = weight + (long)up_k[q] * C32 * C32;
            float s = 0.0f;
            #pragma unroll 8
            for (int cin = 0; cin < C32; ++cin)
                s += inrow[cin] * W[(long)cin * C32 + lane];
            atomicAdd(&out[(long)up_out[q] * C32 + lane], -s);
        }
    }
}

extern "C" void kernel_launch(void* const* d_in, const int* in_sizes, int n_in,
                              void* d_out, int out_size, void* d_ws, size_t ws_size,
                              hipStream_t stream) {
    const float* feat        = (const float*)d_in[0];
    const float* weight      = (const float*)d_in[1];
    const float* bias        = (const float*)d_in[2];
    const int*   pool_in     = (const int*)d_in[3];
    const int*   pool_out    = (const int*)d_in[4];
    const int*   pool_counts = (const int*)d_in[5];
    const int*   up_in       = (const int*)d_in[6];
    const int*   up_out      = (const int*)d_in[7];
    const int*   up_k        = (const int*)d_in[8];

    const long P  = in_sizes[3];
    const long M  = in_sizes[5];
    const int  Q  = in_sizes[6];
    const long NC = in_sizes[0];          // N * 32

    float* out    = (float*)d_out;
    float* pooled = (float*)d_ws;         // M * 32 floats of scratch

    const int BT = 256;

    // 0) zero pooled accumulator (ws is poisoned between calls)
    {
        long n = M * C32;
        int blocks = (int)((n + BT - 1) / BT);
        if (blocks > 65535) blocks = 65535;
        zero_f32_kernel<<<blocks, BT, 0, stream>>>(pooled, n);
    }
    // 1) pool scatter-add
    {
        long n = P * C32;
        pool_scatter_kernel<<<(int)((n + BT - 1) / BT), BT, 0, stream>>>(
            feat, pool_in, pool_out, pooled, n);
    }
    // 2) divide by counts
    {
        long n = M * C32;
        pool_div_kernel<<<(int)((n + BT - 1) / BT), BT, 0, stream>>>(
            pooled, pool_counts, n);
    }
    // 3) out = feat - bias
    {
        init_out_kernel<<<(int)((NC + BT - 1) / BT), BT, 0, stream>>>(
            feat, bias, out, NC);
    }
    // 4) WMMA transposed-conv, scatter-subtract into out
    {
        int tiles  = (Q + 15) / 16;
        int blocks = (tiles + 7) / 8;     // 8 waves (tiles) per 256-thread block
        upconv_wmma_kernel<<<blocks, BT, 0, stream>>>(
            pooled, weight, up_in, up_out, up_k, out, Q);
    }
}